// MGAT_66013647339807
// MI455X (gfx1250) — compile-verified
//
#include <hip/hip_runtime.h>
#include <math.h>

#define NNODES 50000

typedef __attribute__((ext_vector_type(2))) float v2f;
typedef __attribute__((ext_vector_type(8))) float v8f;

__device__ __forceinline__ float lrelu01(float x) { return x >= 0.0f ? x : 0.01f * x; }

// Monotonic order-preserving float -> uint encoding (atomicMax over floats incl.
// negatives). Init value 0 is below every encoded real.
__device__ __forceinline__ unsigned enc_f32(float f) {
    unsigned u = __float_as_uint(f);
    return (u & 0x80000000u) ? ~u : (u | 0x80000000u);
}
__device__ __forceinline__ float dec_f32(unsigned k) {
    return (k & 0x80000000u) ? __uint_as_float(k & 0x7FFFFFFFu) : __uint_as_float(~k);
}

__device__ __forceinline__ float wave_sum(float v) {
#pragma unroll
    for (int off = 16; off > 0; off >>= 1) v += __shfl_xor(v, off, 32);
    return v;
}

// B-fragment load: element (ka, n) and (ka+1, n) of the K x Ncols operand.
// BTRANS=1: B(k,n) = Bm[n*ldb+k]  (two adjacent floats -> v2f load)
// BTRANS=0: B(k,n) = Bm[k*ldb+n]  (two strided floats)
template <int BTRANS>
__device__ __forceinline__ v2f loadB(const float* __restrict__ p, int ldb) {
    v2f b;
    if (BTRANS) {
        b = *(const v2f*)p;
    } else {
        b.x = p[0];
        b.y = p[ldb];
    }
    return b;
}

__device__ __forceinline__ v8f wmma4(v2f a, v2f b, v8f c) {
    return __builtin_amdgcn_wmma_f32_16x16x4_f32(false, a, false, b, (short)0, c,
                                                 false, false);
}

__device__ __forceinline__ void epilogue_tile(v8f acc, int m_base, int half, int n,
                                              int Ncols,
                                              const float* __restrict__ bias,
                                              const float* __restrict__ addend,
                                              const float* __restrict__ postadd,
                                              int act, float* __restrict__ D) {
    if (n < Ncols) {
        const float bv = bias ? bias[n] : 0.0f;
#pragma unroll
        for (int j = 0; j < 8; ++j) {
            const int m = m_base + j + half * 8;  // C/D layout: VGPR j <-> M row
            const long idx = (long)m * Ncols + n;
            float v = acc[j] + bv;
            if (addend) v += addend[idx];
            if (act == 1)      v = tanhf(v);
            else if (act == 2) v = lrelu01(v);
            if (postadd) v += postadd[idx];
            D[idx] = v;
        }
    }
}

// ---------------------------------------------------------------------------
// D = act(A @ B + bias + addend) + postadd
//   A: [M,K] row-major, M multiple of 16, K multiple of 4.
//   act: 0 none, 1 tanh, 2 leaky_relu(0.01)
// One wave computes a 16x64 strip with V_WMMA_F32_16X16X4_F32; A fragment is
// shared across the 4 column tiles; fast path is branch-free and software-
// pipelined (loads for k+4 issued before the 4 WMMAs of k).
// ---------------------------------------------------------------------------
template <int BTRANS>
__global__ void gemm_wmma_f32(const float* __restrict__ A, const float* __restrict__ Bm,
                              float* __restrict__ D,
                              int M, int K, int Ncols, int ldb,
                              const float* __restrict__ bias,
                              const float* __restrict__ addend,
                              const float* __restrict__ postadd,
                              int act) {
    const int lane  = threadIdx.x & 31;
    const int wave  = threadIdx.x >> 5;
    const int mtile = blockIdx.x * 8 + wave;
    if (mtile >= (M >> 4)) return;          // wave-uniform; EXEC stays full below
    const int m_base = mtile << 4;
    const int half   = lane >> 4;           // 0: K{0,1}, rows 0..7 ; 1: K{2,3}, rows 8..15
    const int l16    = lane & 15;
    const float* ap0 = A + (long)(m_base + l16) * (long)K + half * 2;
    const long bstep = BTRANS ? 4 : (long)4 * ldb;

    for (int nb = 0; nb < Ncols; nb += 64) {
        v8f acc0 = {}, acc1 = {}, acc2 = {}, acc3 = {};

        if (nb + 64 <= Ncols) {
            // ---------------- fast path: all 4 tiles fully in range ----------
            const float* ap = ap0;
            const float* p0, *p1, *p2, *p3;
            if (BTRANS) {
                p0 = Bm + (long)(nb + 0 * 16 + l16) * ldb + half * 2;
                p1 = Bm + (long)(nb + 1 * 16 + l16) * ldb + half * 2;
                p2 = Bm + (long)(nb + 2 * 16 + l16) * ldb + half * 2;
                p3 = Bm + (long)(nb + 3 * 16 + l16) * ldb + half * 2;
            } else {
                p0 = Bm + (long)(half * 2) * ldb + (nb + 0 * 16 + l16);
                p1 = Bm + (long)(half * 2) * ldb + (nb + 1 * 16 + l16);
                p2 = Bm + (long)(half * 2) * ldb + (nb + 2 * 16 + l16);
                p3 = Bm + (long)(half * 2) * ldb + (nb + 3 * 16 + l16);
            }
            v2f a_c  = *(const v2f*)ap;
            v2f b0c = loadB<BTRANS>(p0, ldb);
            v2f b1c = loadB<BTRANS>(p1, ldb);
            v2f b2c = loadB<BTRANS>(p2, ldb);
            v2f b3c = loadB<BTRANS>(p3, ldb);
            for (int k0 = 0; k0 + 4 < K; k0 += 4) {
                ap += 4; p0 += bstep; p1 += bstep; p2 += bstep; p3 += bstep;
                v2f a_n  = *(const v2f*)ap;
                v2f b0n = loadB<BTRANS>(p0, ldb);
                v2f b1n = loadB<BTRANS>(p1, ldb);
                v2f b2n = loadB<BTRANS>(p2, ldb);
                v2f b3n = loadB<BTRANS>(p3, ldb);
                acc0 = wmma4(a_c, b0c, acc0);
                acc1 = wmma4(a_c, b1c, acc1);
                acc2 = wmma4(a_c, b2c, acc2);
                acc3 = wmma4(a_c, b3c, acc3);
                a_c = a_n; b0c = b0n; b1c = b1n; b2c = b2n; b3c = b3n;
            }
            acc0 = wmma4(a_c, b0c, acc0);
            acc1 = wmma4(a_c, b1c, acc1);
            acc2 = wmma4(a_c, b2c, acc2);
            acc3 = wmma4(a_c, b3c, acc3);
        } else {
            // ---------------- masked tail: clamp address, multiply by 0/1 ----
            const int Nc1 = Ncols - 1;
            const int n0 = nb + 0 * 16 + l16, n1 = nb + 1 * 16 + l16;
            const int n2 = nb + 2 * 16 + l16, n3 = nb + 3 * 16 + l16;
            const float m0 = (n0 < Ncols) ? 1.0f : 0.0f;
            const float m1 = (n1 < Ncols) ? 1.0f : 0.0f;
            const float m2 = (n2 < Ncols) ? 1.0f : 0.0f;
            const float m3 = (n3 < Ncols) ? 1.0f : 0.0f;
            const int c0 = min(n0, Nc1), c1 = min(n1, Nc1);
            const int c2 = min(n2, Nc1), c3 = min(n3, Nc1);
            const float* ap = ap0;
            const float* p0, *p1, *p2, *p3;
            if (BTRANS) {
                p0 = Bm + (long)c0 * ldb + half * 2;
                p1 = Bm + (long)c1 * ldb + half * 2;
                p2 = Bm + (long)c2 * ldb + half * 2;
                p3 = Bm + (long)c3 * ldb + half * 2;
            } else {
                p0 = Bm + (long)(half * 2) * ldb + c0;
                p1 = Bm + (long)(half * 2) * ldb + c1;
                p2 = Bm + (long)(half * 2) * ldb + c2;
                p3 = Bm + (long)(half * 2) * ldb + c3;
            }
            for (int k0 = 0; k0 < K; k0 += 4) {
                v2f a = *(const v2f*)ap;
                v2f b0 = loadB<BTRANS>(p0, ldb);
                v2f b1 = loadB<BTRANS>(p1, ldb);
                v2f b2 = loadB<BTRANS>(p2, ldb);
                v2f b3 = loadB<BTRANS>(p3, ldb);
                b0.x *= m0; b0.y *= m0;
                b1.x *= m1; b1.y *= m1;
                b2.x *= m2; b2.y *= m2;
                b3.x *= m3; b3.y *= m3;
                acc0 = wmma4(a, b0, acc0);
                acc1 = wmma4(a, b1, acc1);
                acc2 = wmma4(a, b2, acc2);
                acc3 = wmma4(a, b3, acc3);
                ap += 4; p0 += bstep; p1 += bstep; p2 += bstep; p3 += bstep;
            }
        }

        epilogue_tile(acc0, m_base, half, nb + 0 * 16 + l16, Ncols, bias, addend, postadd, act, D);
        epilogue_tile(acc1, m_base, half, nb + 1 * 16 + l16, Ncols, bias, addend, postadd, act, D);
        epilogue_tile(acc2, m_base, half, nb + 2 * 16 + l16, Ncols, bias, addend, postadd, act, D);
        epilogue_tile(acc3, m_base, half, nb + 3 * 16 + l16, Ncols, bias, addend, postadd, act, D);
    }
}

// Row-wise L2 normalize in place. One wave per row.
__global__ void l2norm_rows(float* __restrict__ X, int n, int d) {
    const int lane = threadIdx.x & 31;
    const int row  = (int)((blockIdx.x * blockDim.x + threadIdx.x) >> 5);
    if (row >= n) return;
    float s = 0.0f;
    for (int c = lane; c < d; c += 32) { float v = X[(long)row * d + c]; s += v * v; }
    s = wave_sum(s);
    const float inv = 1.0f / fmaxf(sqrtf(s), 1e-12f);
    for (int c = lane; c < d; c += 32) X[(long)row * d + c] *= inv;
}

// h = lrelu(l2norm(agg_row + bias)). One wave per row, in place.
__global__ void gat_finish(float* __restrict__ AGG, const float* __restrict__ bias,
                           int n, int d) {
    const int lane = threadIdx.x & 31;
    const int row  = (int)((blockIdx.x * blockDim.x + threadIdx.x) >> 5);
    if (row >= n) return;
    float s = 0.0f;
    for (int c = lane; c < d; c += 32) {
        float v = AGG[(long)row * d + c] + bias[c];
        s += v * v;
    }
    s = wave_sum(s);
    const float inv = 1.0f / fmaxf(sqrtf(s), 1e-12f);
    for (int c = lane; c < d; c += 32) {
        float v = (AGG[(long)row * d + c] + bias[c]) * inv;
        AGG[(long)row * d + c] = lrelu01(v);
    }
}

__global__ void fill_f32(float* __restrict__ p, float v, long n) {
    long i = (long)blockIdx.x * blockDim.x + threadIdx.x;
    if (i < n) p[i] = v;
}

// Pass 1: gated score t = ip*sigmoid(ip); segment max via atomicMax on encoded uint.
__global__ void edge_score_max(const float* __restrict__ XW, const int* __restrict__ src,
                               const int* __restrict__ dst, int e_cnt, int d,
                               float* __restrict__ et, unsigned* __restrict__ mkey) {
    const int lane = threadIdx.x & 31;
    const int e = (int)((blockIdx.x * blockDim.x + threadIdx.x) >> 5);
    if (e >= e_cnt) return;
    const int s = src[e], t = dst[e];
    float p = 0.0f;
    for (int c = lane; c < d; c += 32)
        p += XW[(long)s * d + c] * XW[(long)t * d + c];
    p = wave_sum(p);
    if (lane == 0) {
        const float sig = 1.0f / (1.0f + expf(-p));
        const float sc = p * sig;
        et[e] = sc;
        atomicMax(&mkey[t], enc_f32(sc));
    }
}

// Pass 2: e = exp(t - m[dst]); den[dst] += e. Thread per edge.
__global__ void edge_exp_sum(float* __restrict__ et, const unsigned* __restrict__ mkey,
                             const int* __restrict__ dst, int e_cnt,
                             float* __restrict__ den) {
    const int e = blockIdx.x * blockDim.x + threadIdx.x;
    if (e >= e_cnt) return;
    const int t = dst[e];
    const float ex = expf(et[e] - dec_f32(mkey[t]));
    et[e] = ex;
    atomicAdd(&den[t], ex);
}

// Pass 3: agg[dst] += xw[src] * alpha. One wave per edge.
__global__ void edge_aggregate(const float* __restrict__ XW, const float* __restrict__ et,
                               const float* __restrict__ den, const int* __restrict__ src,
                               const int* __restrict__ dst, int e_cnt, int d,
                               float* __restrict__ agg) {
    const int lane = threadIdx.x & 31;
    const int e = (int)((blockIdx.x * blockDim.x + threadIdx.x) >> 5);
    if (e >= e_cnt) return;
    const int s = src[e], t = dst[e];
    const float alpha = et[e] / (den[t] + 1e-16f);
    for (int c = lane; c < d; c += 32)
        atomicAdd(&agg[(long)t * d + c], XW[(long)s * d + c] * alpha);
}

// rep[n,c] = 0.5*(v+t) of concat(x1,x2); BPR scores. One wave per batch element.
__global__ void score_kernel(const float* __restrict__ x1v, const float* __restrict__ x2v,
                             const float* __restrict__ x1t, const float* __restrict__ x2t,
                             const int* __restrict__ user, const int* __restrict__ pos,
                             const int* __restrict__ neg, float* __restrict__ out, int batch) {
    const int lane = threadIdx.x & 31;
    const int b = (int)((blockIdx.x * blockDim.x + threadIdx.x) >> 5);
    if (b >= batch) return;
    const int u = user[b], p = pos[b], g = neg[b];
    float sp = 0.0f, sn = 0.0f;
    for (int c = lane; c < 128; c += 32) {
        float uc, pc, nc;
        if (c < 64) {
            uc = 0.5f * (x1v[(long)u * 64 + c] + x1t[(long)u * 64 + c]);
            pc = 0.5f * (x1v[(long)p * 64 + c] + x1t[(long)p * 64 + c]);
            nc = 0.5f * (x1v[(long)g * 64 + c] + x1t[(long)g * 64 + c]);
        } else {
            const int cc = c - 64;
            uc = 0.5f * (x2v[(long)u * 64 + cc] + x2t[(long)u * 64 + cc]);
            pc = 0.5f * (x2v[(long)p * 64 + cc] + x2t[(long)p * 64 + cc]);
            nc = 0.5f * (x2v[(long)g * 64 + cc] + x2t[(long)g * 64 + cc]);
        }
        sp += uc * pc;
        sn += uc * nc;
    }
    sp = wave_sum(sp);
    sn = wave_sum(sn);
    if (lane == 0) { out[b] = sp; out[batch + b] = sn; }
}

// ---------------------------------------------------------------------------
// Host-side pipeline
// ---------------------------------------------------------------------------
static void run_gat_layer(const float* Xin, int DL, const float* cw, const float* cb,
                          const int* src, const int* dst, int E,
                          float* xw, float* agg, float* et, unsigned* mkey, float* den,
                          hipStream_t stream) {
    const int N = NNODES;
    const int mblocks = (N / 16 + 7) / 8;
    // xw = Xin @ cw   (B not transposed)
    gemm_wmma_f32<0><<<mblocks, 256, 0, stream>>>(Xin, cw, xw, N, DL, DL, DL,
                                                  nullptr, nullptr, nullptr, 0);
    fill_f32<<<(N + 255) / 256, 256, 0, stream>>>((float*)mkey, 0.0f, (long)N);
    fill_f32<<<(N + 255) / 256, 256, 0, stream>>>(den, 0.0f, (long)N);
    fill_f32<<<((long)N * DL + 255) / 256, 256, 0, stream>>>(agg, 0.0f, (long)N * DL);
    edge_score_max<<<(E + 7) / 8, 256, 0, stream>>>(xw, src, dst, E, DL, et, mkey);
    edge_exp_sum<<<(E + 255) / 256, 256, 0, stream>>>(et, mkey, dst, E, den);
    edge_aggregate<<<(E + 7) / 8, 256, 0, stream>>>(xw, et, den, src, dst, E, DL, agg);
    gat_finish<<<(N + 7) / 8, 256, 0, stream>>>(agg, cb, N, DL);  // agg now holds h
}

static void run_modality(const float* feat, int DF, int DL,
                         const float* mlp_w, const float* mlp_b,
                         const float* c1_w, const float* c1_b,
                         const float* l1_w, const float* l1_b,
                         const float* g1_w, const float* g1_b,
                         const float* c2_w, const float* c2_b,
                         const float* l2_w, const float* l2_b,
                         const float* g2_w, const float* g2_b,
                         const float* id_emb, const int* src, const int* dst, int E,
                         float* bufA, float* bufB, float* bufC, float* et,
                         unsigned* mkey, float* den, float* t64,
                         float* x1, float* x2, hipStream_t stream) {
    const int N = NNODES;
    const int mblocks = (N / 16 + 7) / 8;
    // x = l2(tanh(feat @ mlp_w^T + mlp_b))          -> bufA [N, DL]
    gemm_wmma_f32<1><<<mblocks, 256, 0, stream>>>(feat, mlp_w, bufA, N, DF, DL, DF,
                                                  mlp_b, nullptr, nullptr, 1);
    l2norm_rows<<<(N + 7) / 8, 256, 0, stream>>>(bufA, N, DL);
    // h = lrelu(gat1(x))                            -> bufC [N, DL]
    run_gat_layer(bufA, DL, c1_w, c1_b, src, dst, E, bufB, bufC, et, mkey, den, stream);
    // x_hat = lrelu(x @ l1^T + b) + id              -> t64 [N, 64]
    gemm_wmma_f32<1><<<mblocks, 256, 0, stream>>>(bufA, l1_w, t64, N, DL, 64, DL,
                                                  l1_b, nullptr, id_emb, 2);
    // x1 = lrelu(h @ g1^T + b + x_hat)              -> x1 [N, 64]
    gemm_wmma_f32<1><<<mblocks, 256, 0, stream>>>(bufC, g1_w, x1, N, DL, 64, DL,
                                                  g1_b, t64, nullptr, 2);
    // h2 = lrelu(gat2(x1))                          -> bufC [N, 64]
    run_gat_layer(x1, 64, c2_w, c2_b, src, dst, E, bufB, bufC, et, mkey, den, stream);
    // x_hat2 = lrelu(x1 @ l2^T + b) + id            -> t64
    gemm_wmma_f32<1><<<mblocks, 256, 0, stream>>>(x1, l2_w, t64, N, 64, 64, 64,
                                                  l2_b, nullptr, id_emb, 2);
    // x2 = lrelu(h2 @ g2^T + b + x_hat2)            -> x2 [N, 64]
    gemm_wmma_f32<1><<<mblocks, 256, 0, stream>>>(bufC, g2_w, x2, N, 64, 64, 64,
                                                  g2_b, t64, nullptr, 2);
}

extern "C" void kernel_launch(void* const* d_in, const int* in_sizes, int n_in,
                              void* d_out, int out_size, void* d_ws, size_t ws_size,
                              hipStream_t stream) {
    (void)n_in; (void)out_size; (void)ws_size;
    const int N = NNODES;
    auto F = [&](int i) { return (const float*)d_in[i]; };

    const float* v_feat = F(0);
    const float* t_feat = F(1);
    const float* id_emb = F(2);
    const int* ei   = (const int*)d_in[31];
    const int  E    = in_sizes[31] / 2;      // 400000
    const int* src  = ei;
    const int* dst  = ei + E;
    const int* user = (const int*)d_in[32];
    const int* pos  = (const int*)d_in[33];
    const int* neg  = (const int*)d_in[34];
    const int  batch = in_sizes[32];

    // Workspace carve-up (floats, 256B aligned chunks)
    float* ws = (float*)d_ws;
    size_t off = 0;
    auto alloc = [&](size_t n) {
        float* p = ws + off;
        off += (n + 63) & ~(size_t)63;
        return p;
    };
    float*    bufA = alloc((size_t)N * 256);
    float*    bufB = alloc((size_t)N * 256);
    float*    bufC = alloc((size_t)N * 256);
    float*    et   = alloc((size_t)E);
    unsigned* mkey = (unsigned*)alloc((size_t)N);
    float*    den  = alloc((size_t)N);
    float*    t64  = alloc((size_t)N * 64);
    float*    x1v  = alloc((size_t)N * 64);
    float*    x2v  = alloc((size_t)N * 64);
    float*    x1t  = alloc((size_t)N * 64);
    float*    x2t  = alloc((size_t)N * 64);

    // Visual branch: DF=2048, DL=256 (inputs 3..16)
    run_modality(v_feat, 2048, 256,
                 F(3), F(4), F(5), F(6), F(7), F(8), F(9), F(10),
                 F(11), F(12), F(13), F(14), F(15), F(16),
                 id_emb, src, dst, E,
                 bufA, bufB, bufC, et, mkey, den, t64, x1v, x2v, stream);
    // Text branch: DF=768, DL=100 (inputs 17..30)
    run_modality(t_feat, 768, 100,
                 F(17), F(18), F(19), F(20), F(21), F(22), F(23), F(24),
                 F(25), F(26), F(27), F(28), F(29), F(30),
                 id_emb, src, dst, E,
                 bufA, bufB, bufC, et, mkey, den, t64, x1t, x2t, stream);

    score_kernel<<<(batch + 7) / 8, 256, 0, stream>>>(x1v, x2v, x1t, x2t,
                                                      user, pos, neg,
                                                      (float*)d_out, batch);
}